// TraceCaptionContrastiveModel_13520557048021
// MI455X (gfx1250) — compile-verified
//
#include <hip/hip_runtime.h>
#include <hip/hip_bf16.h>

typedef __attribute__((ext_vector_type(2))) float v2f;
typedef __attribute__((ext_vector_type(8))) float v8f;

#define D 768
#define S 8
#define N_VIS 100
#define L_CAP 512
#define L_TRACE 412
#define B 64
#define K2 (2 * D)   // 1536

// ---------------------------------------------------------------------------
// Kernel 1: caption segment means.
// grid = B blocks, block = 768 threads (one thread per feature dim).
// ---------------------------------------------------------------------------
__global__ __launch_bounds__(D) void caption_segmean_kernel(
    const float* __restrict__ caption_feat,   // [B, 512, 768]
    const int* __restrict__ caption_mask,     // [B, 513]
    float* __restrict__ caption_out)          // [B, 8, 768]
{
    __shared__ int smask[L_CAP];
    const int b = blockIdx.x;
    const int d = threadIdx.x;

    // cm = caption_mask[b, 1:513]
    for (int t = d; t < L_CAP; t += D)
        smask[t] = caption_mask[b * (L_CAP + 1) + 1 + t];
    __syncthreads();

    float acc[S];
    int cnt[S];
#pragma unroll
    for (int s = 0; s < S; ++s) { acc[s] = 0.0f; cnt[s] = 0; }

    const float* fp = caption_feat + (size_t)b * L_CAP * D + d;
    for (int t = 0; t < L_CAP; ++t) {
        const int m = smask[t];
        const float v = fp[(size_t)t * D];
#pragma unroll
        for (int s = 0; s < S; ++s) {
            const bool hit = (m == s + 1);
            acc[s] += hit ? v : 0.0f;
            cnt[s] += hit ? 1 : 0;
        }
    }

    float* op = caption_out + ((size_t)b * S) * D + d;
#pragma unroll
    for (int s = 0; s < S; ++s)
        op[(size_t)s * D] = (cnt[s] > 0) ? (acc[s] / (float)cnt[s]) : 0.0f;
}

// ---------------------------------------------------------------------------
// Kernel 2: vision mean + trace segment means -> pack vt [B*S, 1536] and
// present flags [B*S] into workspace.
// grid = B blocks, block = 768 threads.
// ---------------------------------------------------------------------------
__global__ __launch_bounds__(D) void trace_pack_kernel(
    const float* __restrict__ vision_trace_feat, // [B, 512, 768]
    const int* __restrict__ vision_trace_mask,   // [B, 412]
    float* __restrict__ vt,                      // [B*S, 1536]
    float* __restrict__ present)                 // [B*S]
{
    __shared__ int smask[L_TRACE];
    const int b = blockIdx.x;
    const int d = threadIdx.x;

    for (int t = d; t < L_TRACE; t += D)
        smask[t] = vision_trace_mask[b * L_TRACE + t];
    __syncthreads();

    const float* fp = vision_trace_feat + (size_t)b * (N_VIS + L_TRACE) * D + d;

    // vision mean over first 100 tokens
    float vacc = 0.0f;
    for (int t = 0; t < N_VIS; ++t)
        vacc += fp[(size_t)t * D];
    const float vmean = vacc * (1.0f / (float)N_VIS);

    // trace segment sums over tokens 100..511
    float acc[S];
    int cnt[S];
#pragma unroll
    for (int s = 0; s < S; ++s) { acc[s] = 0.0f; cnt[s] = 0; }

    const float* tp = fp + (size_t)N_VIS * D;
    for (int t = 0; t < L_TRACE; ++t) {
        const int m = smask[t];
        const float v = tp[(size_t)t * D];
#pragma unroll
        for (int s = 0; s < S; ++s) {
            const bool hit = (m == s + 1);
            acc[s] += hit ? v : 0.0f;
            cnt[s] += hit ? 1 : 0;
        }
    }

#pragma unroll
    for (int s = 0; s < S; ++s) {
        const int row = b * S + s;
        const float cntf = (cnt[s] > 0) ? (float)cnt[s] : 1.0f;
        vt[(size_t)row * K2 + d]     = acc[s] / cntf;   // t_means half
        vt[(size_t)row * K2 + D + d] = vmean;           // broadcast vision half
    }
    if (d < S)
        present[b * S + d] = (cnt[d] > 0) ? 1.0f : 0.0f;
}

// ---------------------------------------------------------------------------
// Kernel 3: WMMA fp32 GEMM: out[m, n] = (sum_k vt[m,k] * W[k,n] + bias[n]) *
// present[m].  M = 512, K = 1536, N = 768.
// 256 threads = 8 waves per block; one 16x16 output tile per wave.
// Tiles: (512/16) * (768/16) = 32 * 48 = 1536 -> 192 blocks.
// Uses V_WMMA_F32_16X16X4_F32 (K=4 per op, 384 ops per tile).
// ---------------------------------------------------------------------------
__global__ __launch_bounds__(256) void vt_gemm_wmma_kernel(
    const float* __restrict__ vt,       // [512, 1536]
    const float* __restrict__ W,        // [1536, 768] row-major
    const float* __restrict__ bias,     // [768]
    const float* __restrict__ present,  // [512]
    float* __restrict__ out)            // [512, 768] (vt_out)
{
    const int NT_N = D / 16;            // 48 tiles along N
    const int wave = threadIdx.x >> 5;
    const int lane = threadIdx.x & 31;
    const int tile = blockIdx.x * 8 + wave;
    const int m0 = (tile / NT_N) << 4;
    const int n0 = (tile % NT_N) << 4;

    const int mn = lane & 15;           // A-row (m) / B-col (n) within tile
    const int khalf = lane >> 4;        // 0: K={0,1}  1: K={2,3} per wmma step

    // A fragment source: row (m0+mn) of vt; B fragment source: col (n0+mn) of W
    const float* aRow = vt + (size_t)(m0 + mn) * K2;
    const float* bCol = W + n0 + mn;

    v8f c = {};
    for (int k = 0; k < K2; k += 16) {
#pragma unroll
        for (int kk = 0; kk < 16; kk += 4) {
            const int kb = k + kk + 2 * khalf;
            // A 16x4 layout: lanes 0-15 -> K=0(V0),1(V1); lanes 16-31 -> K=2,3
            v2f a;
            a.x = aRow[kb];
            a.y = aRow[kb + 1];
            // B 4x16 layout mirrored: lanes 0-15 -> K=0,1 rows; 16-31 -> K=2,3
            v2f bb;
            bb.x = bCol[(size_t)kb * D];
            bb.y = bCol[(size_t)(kb + 1) * D];
            c = __builtin_amdgcn_wmma_f32_16x16x4_f32(
                /*neg_a=*/false, a, /*neg_b=*/false, bb,
                /*c_mod=*/(short)0, c, /*reuse_a=*/false, /*reuse_b=*/false);
        }
    }

    // C/D 16x16 f32 layout: lane = mhalf*16 + n ; VGPR r holds M = r + 8*mhalf
    const int n = lane & 15;
    const int mhalf = lane >> 4;
    const float bval = bias[n0 + n];
#pragma unroll
    for (int r = 0; r < 8; ++r) {
        const int m = m0 + r + 8 * mhalf;
        out[(size_t)m * D + n0 + n] = (c[r] + bval) * present[m];
    }
}

// ---------------------------------------------------------------------------
// Launch
// ---------------------------------------------------------------------------
extern "C" void kernel_launch(void* const* d_in, const int* in_sizes, int n_in,
                              void* d_out, int out_size, void* d_ws, size_t ws_size,
                              hipStream_t stream) {
    const float* vision_trace_feat = (const float*)d_in[0]; // [64,512,768]
    const int*   vision_trace_mask = (const int*)d_in[1];   // [64,412]
    const float* caption_feat      = (const float*)d_in[2]; // [64,512,768]
    const int*   caption_mask      = (const int*)d_in[3];   // [64,513]
    const float* vt_merge_w        = (const float*)d_in[4]; // [1536,768]
    const float* vt_merge_b        = (const float*)d_in[5]; // [768]

    float* caption_out = (float*)d_out;                       // [64,8,768]
    float* vt_out      = (float*)d_out + (size_t)B * S * D;   // [64,8,768]

    // Workspace: vt matrix [512,1536] then present [512]  (~3.15 MB)
    float* vt      = (float*)d_ws;
    float* present = vt + (size_t)B * S * K2;

    caption_segmean_kernel<<<B, D, 0, stream>>>(caption_feat, caption_mask,
                                                caption_out);
    trace_pack_kernel<<<B, D, 0, stream>>>(vision_trace_feat, vision_trace_mask,
                                           vt, present);
    const int tiles = (B * S / 16) * (D / 16);   // 32 * 48 = 1536
    vt_gemm_wmma_kernel<<<tiles / 8, 256, 0, stream>>>(vt, vt_merge_w,
                                                       vt_merge_b, present,
                                                       vt_out);
}